// sparse_conv3d_naive_74483322847782
// MI455X (gfx1250) — compile-verified
//
#include <hip/hip_runtime.h>

// Deformable conv3d (N=2, C=64, D=16, H=32, W=32, K=3^3=27, stride=1, pad=1, dil=1)
// Fused: trilinear sampling tile -> LDS -> V_WMMA_F32_16X16X4_F32 accumulation.
//
// GEMM view: out[64, 32768] = weight[64, 64*27] x sampled[64*27, 32768]
// Block = 128 threads (4 wave32): 16-position tile, each wave a 16x16 (o x p) tile.

typedef __attribute__((ext_vector_type(2))) float v2f;
typedef __attribute__((ext_vector_type(8))) float v8f;

#define C_IN   64
#define C_OUT  64
#define DD     16
#define HH     32
#define WW     32
#define KN     27
#define SPATIAL (DD * HH * WW)   // 16384
#define TILE_P  16

__launch_bounds__(128)
__global__ void dcn3d_wmma_f32_kernel(const float* __restrict__ inp,
                                      const float* __restrict__ off,
                                      const float* __restrict__ wgt,
                                      float* __restrict__ out) {
    // sampled tile: [channel][position], 64*16*4B = 4KB LDS
    __shared__ float smp[C_IN][TILE_P];

    const int tid  = threadIdx.x;          // 0..127
    const int wave = tid >> 5;             // 0..3 -> output-channel tile (16 each)
    const int lane = tid & 31;

    const int p_base  = blockIdx.x * TILE_P;     // global position base
    const int n       = p_base / SPATIAL;        // batch
    const int sp_base = p_base % SPATIAL;        // spatial base (aligned to 16, within one (d,h) row)

    // ---- sampling role: thread = (position-in-tile, 8-channel group) ----
    const int sp_p = tid & 15;             // 0..15 position within tile
    const int cg   = tid >> 4;             // 0..7 channel group (8 channels each)
    const int sp   = sp_base + sp_p;
    const int d0   = sp >> 10;             // / (H*W)
    const int h0   = (sp >> 5) & (HH - 1);
    const int w0   = sp & (WW - 1);

    const float* __restrict__ in_n  = inp + (size_t)n * C_IN * SPATIAL;
    const float* __restrict__ off_n = off + (size_t)n * KN * 3 * SPATIAL + sp;

    // grid_sample index mapping (align_corners=False, see derivation):
    //   ix = (w0 + ox) * W/(W-1) - 0.5   (same for y with H, z with D)
    const float sx = (float)WW / (float)(WW - 1);
    const float sy = (float)HH / (float)(HH - 1);
    const float sz = (float)DD / (float)(DD - 1);

    // ---- WMMA roles ----
    const int mrow  = lane & 15;                 // A row (output channel) / B,C,D column (position)
    const int khalf = (lane < 16) ? 0 : 2;       // K offset for upper lane half (A and B layout)
    const int o_row = wave * 16 + mrow;          // this lane's weight row
    // weight[o][c][k] flat index = (o*C_IN + c)*KN + k
    const float* __restrict__ wrow = wgt + (size_t)o_row * C_IN * KN;

    v8f acc = {};

    for (int k = 0; k < KN; ++k) {
        // ================= Phase 1: trilinear sample 64x16 tile =================
        const float oz = off_n[(k * 3 + 0) * SPATIAL];
        const float oy = off_n[(k * 3 + 1) * SPATIAL];
        const float ox = off_n[(k * 3 + 2) * SPATIAL];

        const float iz = ((float)d0 + oz) * sz - 0.5f;
        const float iy = ((float)h0 + oy) * sy - 0.5f;
        const float ix = ((float)w0 + ox) * sx - 0.5f;

        const float zf = floorf(iz), yf = floorf(iy), xf = floorf(ix);
        const int z0i = (int)zf, y0i = (int)yf, x0i = (int)xf;
        const float fz = iz - zf, fy = iy - yf, fx = ix - xf;
        const float wz0 = 1.0f - fz, wy0 = 1.0f - fy, wx0 = 1.0f - fx;

        float cw[8];
        int   cof[8];
#pragma unroll
        for (int j = 0; j < 8; ++j) {
            const int dz = j >> 2, dy = (j >> 1) & 1, dx = j & 1;
            const int zc = z0i + dz, yc = y0i + dy, xc = x0i + dx;
            const bool v = (zc >= 0) & (zc < DD) & (yc >= 0) & (yc < HH) &
                           (xc >= 0) & (xc < WW);
            const int zi = min(max(zc, 0), DD - 1);
            const int yi = min(max(yc, 0), HH - 1);
            const int xi = min(max(xc, 0), WW - 1);
            const float w3 = (dz ? fz : wz0) * (dy ? fy : wy0) * (dx ? fx : wx0);
            cw[j]  = v ? w3 : 0.0f;
            cof[j] = zi * (HH * WW) + yi * WW + xi;
        }

#pragma unroll
        for (int cc = 0; cc < 8; ++cc) {
            const int c = cg * 8 + cc;
            const float* __restrict__ base = in_n + c * SPATIAL;
            float s = 0.0f;
#pragma unroll
            for (int j = 0; j < 8; ++j)
                s = fmaf(cw[j], base[cof[j]], s);
            smp[c][sp_p] = s;
        }
        __syncthreads();

        // ======== Phase 2: acc += weight_tile(16x64) x smp(64x16), K=4 steps ====
        // A 16x4 f32 layout: lane<16 -> {K0,K1}, lane>=16 -> {K2,K3}, row = lane&15
        // B 4x16  f32 layout: mirrored K-halves, column = lane&15
#pragma unroll
        for (int j = 0; j < 16; ++j) {
            const int cb = 4 * j + khalf;
            v2f a, b;
            a[0] = wrow[(cb + 0) * KN + k];
            a[1] = wrow[(cb + 1) * KN + k];
            b[0] = smp[cb + 0][mrow];
            b[1] = smp[cb + 1][mrow];
            acc = __builtin_amdgcn_wmma_f32_16x16x4_f32(
                /*neg_a=*/false, a, /*neg_b=*/false, b,
                /*c_mod=*/(short)0, acc, /*reuse_a=*/false, /*reuse_b=*/false);
        }
        __syncthreads();   // protect smp before next tap overwrites it
    }

    // ================= Store D: VGPR i -> M = i + (lane<16 ? 0 : 8) =============
    const int moff = (lane < 16) ? 0 : 8;
    float* __restrict__ out_base = out + (size_t)n * C_OUT * SPATIAL + sp_base + mrow;
#pragma unroll
    for (int i = 0; i < 8; ++i) {
        const int o = wave * 16 + i + moff;
        out_base[(size_t)o * SPATIAL] = acc[i];
    }
}

extern "C" void kernel_launch(void* const* d_in, const int* in_sizes, int n_in,
                              void* d_out, int out_size, void* d_ws, size_t ws_size,
                              hipStream_t stream) {
    (void)in_sizes; (void)n_in; (void)d_ws; (void)ws_size; (void)out_size;
    const float* inp = (const float*)d_in[0];   // (2, 64, 16, 32, 32) f32
    const float* off = (const float*)d_in[1];   // (2, 81, 16, 32, 32) f32
    const float* wgt = (const float*)d_in[2];   // (64, 64, 27) f32
    float* out = (float*)d_out;                 // (2, 64, 16, 32, 32) f32

    const int total_p = 2 * SPATIAL;            // 32768 output positions
    const int blocks  = total_p / TILE_P;       // 2048
    dcn3d_wmma_f32_kernel<<<blocks, 128, 0, stream>>>(inp, off, wgt, out);
}